// ConditionalRandomField_89953795048078
// MI455X (gfx1250) — compile-verified
//
#include <hip/hip_runtime.h>

// CRF NLL on gfx1250: forward DP as chained V_WMMA_F32_16X16X4_F32 GEMMs.
// B=512 batches, L=1024 steps, T=48 tags.
// Latency-oriented: 1 barrier/step, reference-column stabilization (no
// cross-lane reductions), raw v_log_f32/v_exp_f32, b128 LDS operand traffic,
// time loop unrolled x2 so ping-pong buffer offsets are immediates.

#define B_ 512
#define L_ 1024
#define T_ 48

#define LOG2E 1.4426950408889634f
#define LN2   0.6931471805599453f

typedef float v2f __attribute__((ext_vector_type(2)));
typedef float v4f __attribute__((ext_vector_type(4)));
typedef float v8f __attribute__((ext_vector_type(8)));

// ---------------------------------------------------------------------------
// Gold path score.
// ---------------------------------------------------------------------------
__global__ __launch_bounds__(256) void crf_gold_kernel(
    const float* __restrict__ feats, const int* __restrict__ tags,
    const int* __restrict__ mask, const float* __restrict__ trans,
    const float* __restrict__ start_s, const float* __restrict__ end_s,
    float* __restrict__ gold)
{
    __shared__ float sred[256];
    __shared__ int   cred[256];
    const int b = blockIdx.x, tid = threadIdx.x;
    float s = 0.f; int cnt = 0;
    for (int t = tid; t < L_; t += 256) {
        const int mt = mask[b * L_ + t];
        if (mt) {
            const int tg = tags[b * L_ + t];
            s += feats[(b * L_ + t) * T_ + tg];
            ++cnt;
            if (t > 0) s += trans[tags[b * L_ + t - 1] * T_ + tg];
        }
    }
    sred[tid] = s; cred[tid] = cnt;
    __syncthreads();
    for (int off = 128; off >= 1; off >>= 1) {
        if (tid < off) { sred[tid] += sred[tid + off]; cred[tid] += cred[tid + off]; }
        __syncthreads();
    }
    if (tid == 0) {
        int last = cred[0] - 1; if (last < 0) last = 0;
        gold[b] = sred[0] + start_s[tags[b * L_]] + end_s[tags[b * L_ + last]];
    }
}

// ---------------------------------------------------------------------------
// Forward DP. One workgroup = 16 batch rows; 3 waves = 3 N-tiles of 16 tags.
// Per step: D[16x16] = exp(alpha - ref)[16x48] x expT[48x16] via 12 chained
// f32 WMMAs; alpha' = ref + ln2*log2(D) + emit.
// Stabilizer ref = alpha_{t-1}[row][0] (bounded distance from the row max,
// so no per-step max reduction is needed).
// ---------------------------------------------------------------------------
#define USTRIDE 52   // floats per row: multiple of 4 (16B), conflict-free banks

__global__ __launch_bounds__(96) void crf_forward_kernel(
    const float* __restrict__ feats, const int* __restrict__ mask,
    const float* __restrict__ trans, const float* __restrict__ start_s,
    const float* __restrict__ end_s, const float* __restrict__ gold,
    float* __restrict__ out)
{
    __shared__ __align__(16) float u_ld[2][16 * USTRIDE]; // ping-pong by t parity
    __shared__ __align__(16) float refc[2][16];           // ref column per row

    const int tid  = threadIdx.x;
    const int wv   = tid >> 5;        // N-tile id (0..2)
    const int lane = tid & 31;
    const int l15  = lane & 15;
    const int kh   = lane >> 4;       // K-half selector (ISA 16x16x4 layout)
    const int b0   = blockIdx.x * 16;
    const int colg = wv * 16 + l15;   // global tag column for B/D operands

    // Permuted u layout: k = kc*4 + khs*2 + i  ->  khs*24 + kc*2 + i
    const int poff = ((colg & 2) >> 1) * 24 + (colg >> 2) * 2 + (colg & 1);

    // B operand: exp(trans) resident in 24 VGPRs.
    // 4x16 f32 B layout: VGPR0 = K{0|2}, VGPR1 = K{1|3} (kh selects +2).
    v2f bm[12];
    #pragma unroll
    for (int kc = 0; kc < 12; ++kc) {
        const int k0 = kc * 4 + 2 * kh;
        v2f b;
        b.x = __builtin_amdgcn_exp2f(trans[(k0 + 0) * T_ + colg] * LOG2E);
        b.y = __builtin_amdgcn_exp2f(trans[(k0 + 1) * T_ + colg] * LOG2E);
        bm[kc] = b;
    }

    const int rowstride = L_ * T_;
    const int ebase = (b0 + 8 * kh) * rowstride + colg;   // D-layout emission base
    const int mbase = (b0 + 8 * kh) * L_;

    // ---- t = 0: alpha0 = feats[:,0,:] + start ----
    v8f cur;
    const float st = start_s[colg];
    #pragma unroll
    for (int r = 0; r < 8; ++r)
        cur[r] = feats[ebase + r * rowstride] + st;

    // publish reference column (alpha0[row][0]) -- wave 0, lanes 0/16 own col 0
    if (wv == 0 && l15 == 0) {
        v4f lo = {cur[0], cur[1], cur[2], cur[3]};
        v4f hi = {cur[4], cur[5], cur[6], cur[7]};
        *(v4f*)&refc[0][8 * kh + 0] = lo;
        *(v4f*)&refc[0][8 * kh + 4] = hi;
    }
    __syncthreads();
    float mreg[8];   // exact value subtracted when the current u was built
    {
        const v4f m0 = *(const v4f*)&refc[0][8 * kh + 0];
        const v4f m1 = *(const v4f*)&refc[0][8 * kh + 4];
        #pragma unroll
        for (int r = 0; r < 8; ++r) {
            const int row = r + 8 * kh;
            const float mn = (r < 4) ? m0[r & 3] : m1[r & 3];
            u_ld[0][row * USTRIDE + poff] =
                __builtin_amdgcn_exp2f((cur[r] - mn) * LOG2E);
            mreg[r] = mn;
        }
    }
    __syncthreads();

    // Double-buffered emissions/mask.
    float e_n[8]; int m_n[8];
    #pragma unroll
    for (int r = 0; r < 8; ++r) {
        e_n[r] = feats[ebase + r * rowstride + T_];
        m_n[r] = mask[mbase + r * L_ + 1];
    }

    // One DP step; RB/WB are compile-time ping-pong buffer indices.
#define STEP(TT, RB, WB)                                                        \
    {                                                                           \
        const int t_ = (TT);                                                    \
        float e_c[8]; int m_c[8];                                               \
        _Pragma("unroll")                                                       \
        for (int r = 0; r < 8; ++r) { e_c[r] = e_n[r]; m_c[r] = m_n[r]; }       \
        if (t_ + 1 < L_) {                                                      \
            _Pragma("unroll")                                                   \
            for (int r = 0; r < 8; ++r) {                                       \
                e_n[r] = feats[ebase + r * rowstride + (t_ + 1) * T_];          \
                m_n[r] = mask[mbase + r * L_ + (t_ + 1)];                       \
            }                                                                   \
        }                                                                       \
        if (t_ + 2 < L_)                                                        \
            __builtin_prefetch(&feats[ebase + (t_ + 2) * T_], 0, 0);            \
        /* A operand: 24 contiguous floats per lane = 6 x b128 */               \
        v4f q[6];                                                               \
        _Pragma("unroll")                                                       \
        for (int j = 0; j < 6; ++j)                                             \
            q[j] = *(const v4f*)&u_ld[RB][l15 * USTRIDE + kh * 24 + j * 4];     \
        /* 12 chained K=4 WMMAs: acc_j = sum_i u_i * expT[i][j] */              \
        v8f acc = {};                                                           \
        _Pragma("unroll")                                                       \
        for (int kc = 0; kc < 12; ++kc) {                                       \
            v2f a;                                                              \
            a.x = q[kc >> 1][(kc & 1) * 2 + 0];                                 \
            a.y = q[kc >> 1][(kc & 1) * 2 + 1];                                 \
            acc = __builtin_amdgcn_wmma_f32_16x16x4_f32(                        \
                false, a, false, bm[kc], (short)0, acc, false, false);          \
        }                                                                       \
        /* alpha' = ref + ln2*log2(acc) + emit, masked vs old alpha */          \
        _Pragma("unroll")                                                       \
        for (int r = 0; r < 8; ++r) {                                           \
            const float na = fmaf(__builtin_amdgcn_logf(acc[r]), LN2,           \
                                  mreg[r] + e_c[r]);                            \
            cur[r] = m_c[r] ? na : cur[r];                                      \
        }                                                                       \
        /* publish new reference column (col 0 lives in wave 0, lanes 0/16) */  \
        if (wv == 0 && l15 == 0) {                                              \
            v4f lo = {cur[0], cur[1], cur[2], cur[3]};                          \
            v4f hi = {cur[4], cur[5], cur[6], cur[7]};                          \
            *(v4f*)&refc[WB][8 * kh + 0] = lo;                                  \
            *(v4f*)&refc[WB][8 * kh + 4] = hi;                                  \
        }                                                                       \
        /* publish u_t stabilized by one-step-stale reference column */         \
        {                                                                       \
            const v4f m0 = *(const v4f*)&refc[RB][8 * kh + 0];                  \
            const v4f m1 = *(const v4f*)&refc[RB][8 * kh + 4];                  \
            _Pragma("unroll")                                                   \
            for (int r = 0; r < 8; ++r) {                                       \
                const int row = r + 8 * kh;                                     \
                const float mn = (r < 4) ? m0[r & 3] : m1[r & 3];               \
                u_ld[WB][row * USTRIDE + poff] =                                \
                    __builtin_amdgcn_exp2f((cur[r] - mn) * LOG2E);              \
                mreg[r] = mn;                                                   \
            }                                                                   \
        }                                                                       \
        __syncthreads();                                                        \
    }

    STEP(1, 0, 1);
    for (int t = 2; t < L_; t += 2) {
        STEP(t, 1, 0);
        STEP(t + 1, 0, 1);
    }
#undef STEP

    // ---- epilogue: out[b] = logsumexp_j(alpha_j + end_j) - gold[b] ----
    #pragma unroll
    for (int r = 0; r < 8; ++r) {
        const int row = r + 8 * kh;
        u_ld[0][row * USTRIDE + colg] = cur[r] + end_s[colg];
    }
    __syncthreads();
    if (tid < 16) {
        float m = -3.402823466e38f;
        for (int j = 0; j < T_; ++j)
            m = fmaxf(m, u_ld[0][tid * USTRIDE + j]);
        float s = 0.f;
        for (int j = 0; j < T_; ++j)
            s += __builtin_amdgcn_exp2f((u_ld[0][tid * USTRIDE + j] - m) * LOG2E);
        out[b0 + tid] = fmaf(__builtin_amdgcn_logf(s), LN2, m) - gold[b0 + tid];
    }
}

// ---------------------------------------------------------------------------
extern "C" void kernel_launch(void* const* d_in, const int* in_sizes, int n_in,
                              void* d_out, int out_size, void* d_ws, size_t ws_size,
                              hipStream_t stream) {
    const float* feats   = (const float*)d_in[0];
    const int*   tags    = (const int*)d_in[1];
    const int*   mask    = (const int*)d_in[2];
    const float* trans   = (const float*)d_in[3];
    const float* start_s = (const float*)d_in[4];
    const float* end_s   = (const float*)d_in[5];
    float* out  = (float*)d_out;
    float* gold = (float*)d_ws;   // 512 floats of scratch

    crf_gold_kernel<<<B_, 256, 0, stream>>>(feats, tags, mask, trans,
                                            start_s, end_s, gold);
    crf_forward_kernel<<<B_ / 16, 96, 0, stream>>>(feats, mask, trans,
                                                   start_s, end_s, gold, out);
}